// ButterflyGatingUnit_25864293056692
// MI455X (gfx1250) — compile-verified
//
#include <hip/hip_runtime.h>
#include <math.h>

// ---------------------------------------------------------------------------
// ButterflyGatingUnit for MI455X (gfx1250, wave32).
// All GEMM-like work (5 convs + QK attention + PV-as-dynamic-conv) runs on
// V_WMMA_F32_16X16X4_F32. Inner loops are restructured so the hot path is
// loads + v_wmma only: k-decomposition and bounds checks hoisted, zero-page
// pointer select for padding, weights pre-transposed for b64 A-loads, and
// two N-tiles per wave sharing the A fragment (2 independent WMMA chains).
// ---------------------------------------------------------------------------

typedef float v2f __attribute__((ext_vector_type(2)));
typedef float v8f __attribute__((ext_vector_type(8)));

constexpr int cB = 8;
constexpr int cC = 96;
constexpr int cH = 96;
constexpr int cW = 96;
constexpr int cHW = cH * cW;              // 9216
constexpr int cNPIX = cB * cHW;           // 73728
constexpr int cM = cC * 9;                // 864 (softmax row length)
constexpr int cZB = 2 * cC * cHW + 16;    // zero page: covers Cin=192 planes
constexpr float cSCALE = 0.03402069087198859f; // 1/sqrt(864)

__device__ __forceinline__ v8f wmma_f32(v2f a, v2f b, v8f c) {
  // (neg_a, A, neg_b, B, c_mod, C, reuse_a, reuse_b)
  return __builtin_amdgcn_wmma_f32_16x16x4_f32(false, a, false, b, (short)0, c,
                                               false, false);
}

// ---------------------------------------------------------------------------
// Weight reorder: wt[b][m][r*Cin + ci] = w[b][m][ci*9 + r].
// Makes the per-lane A fragment (2 consecutive K slots) contiguous -> b64.
// nb = 1 for static conv weights, 8 for the per-batch attention weights.
// ---------------------------------------------------------------------------
__global__ __launch_bounds__(256) void reorder_w(const float* __restrict__ w,
                                                 float* __restrict__ wt,
                                                 int Cin, int nb) {
  const int Kdim = Cin * 9;
  const size_t total = (size_t)nb * cC * Kdim;
  const size_t idx = (size_t)blockIdx.x * 256 + threadIdx.x;
  if (idx >= total) return;
  const int t = (int)(idx % Kdim);
  const size_t bm = idx / Kdim;
  const int r = t / Cin;
  const int ci = t - r * Cin;
  wt[idx] = w[bm * Kdim + (size_t)ci * 9 + r];
}

__global__ __launch_bounds__(256) void fill_zero(float* __restrict__ p, int n) {
  const int i = blockIdx.x * 256 + threadIdx.x;
  if (i < n) p[i] = 0.0f;
}

// ---------------------------------------------------------------------------
// Implicit-GEMM 3x3 same-pad conv: out[b,co,h,w] = sum_{ci,kh,kw} in*w.
//   in : [B, Cin, 96, 96]        wt : [Cout=96][r=9][Cin] (pre-transposed)
//   out: [B, 96, 96, 96]         zb : zero page for padding taps
// Each wave: 16(co) x 32(pixels), two v8f accumulators sharing A.
// K loop: r outer (bounds/pointer setup), ci inner (pure load+wmma).
// Tiles: 6 M-tiles * 2304 N-tiles = 13824 waves -> 1728 blocks of 8 waves.
// ---------------------------------------------------------------------------
__global__ __launch_bounds__(256) void conv3x3_wmma(
    const float* __restrict__ in, const float* __restrict__ wt,
    float* __restrict__ out, const float* __restrict__ zb, int Cin,
    int perBatchW /*0 or 96*Kdim*/) {
  const int Kdim = Cin * 9;
  const int wave = threadIdx.x >> 5;
  const int lane = threadIdx.x & 31;
  const int tile = blockIdx.x * 8 + wave;
  const int mt = tile % 6;
  const int nt = tile / 6;                  // 32-pixel tiles, row-contained
  const int n0 = nt << 5;
  const int b  = n0 / cHW;
  const int hw = n0 - b * cHW;
  const int h  = hw / cW;
  const int w0 = hw - h * cW;               // in {0,32,64}
  const int hl  = lane & 15;
  const int kh2 = (lane >> 4) << 1;         // K slot offset 0 or 2

  const float* wp = wt + (size_t)(perBatchW ? b * perBatchW : 0) +
                    (size_t)((mt << 4) + hl) * Kdim + kh2;
  const float* inb = in + (size_t)b * Cin * cHW;

  v8f acc0 = {}, acc1 = {};
#pragma unroll
  for (int r = 0; r < 9; ++r) {
    const int dh = r / 3;                   // constant after unroll
    const int dw = r - dh * 3;
    const int hh = h + dh - 1;
    const int ww0 = w0 + hl + dw - 1;
    const int ww1 = ww0 + 16;
    const bool hok = (unsigned)hh < (unsigned)cH;
    const size_t row = (size_t)hh * cW;
    const float* p0 = (hok && (unsigned)ww0 < (unsigned)cW) ? inb + row + ww0 : zb;
    const float* p1 = (hok && (unsigned)ww1 < (unsigned)cW) ? inb + row + ww1 : zb;
    const float* wr = wp + r * Cin;
    __builtin_prefetch(p0 + 4 * cHW, 0, 0); // stream hint -> global_prefetch_b8
#pragma unroll 4
    for (int ci = 0; ci < Cin; ci += 4) {
      const v2f a = *(const v2f*)(wr + ci);
      const size_t o = (size_t)(ci + kh2) * cHW;
      v2f b0, b1;
      b0[0] = p0[o];
      b0[1] = p0[o + cHW];
      b1[0] = p1[o];
      b1[1] = p1[o + cHW];
      acc0 = wmma_f32(a, b0, acc0);
      acc1 = wmma_f32(a, b1, acc1);
    }
  }

  // D layout: N = lane%16, M = rr + 8*(lane/16)
  const int mrow = (mt << 4) + ((lane >> 4) << 3);
  float* op = out + (((size_t)b * cC + mrow) * cH + h) * cW + w0 + hl;
#pragma unroll
  for (int rr = 0; rr < 8; ++rr) {
    op[(size_t)rr * cHW] = acc0[rr];
    op[(size_t)rr * cHW + 16] = acc1[rr];
  }
}

// ---------------------------------------------------------------------------
// QK attention: for each (b, p=kh*3+kw): attn[c1,c2] = sum_l q_p[c1,l]*k_p[c2,l]
// l = li*32+lj over the 32x32 grid of non-overlapping 3x3 cells.
// Each wave: 16(c1) x 32(c2) tile, A shared by two accumulators.
// 72 (b,p) * 6 mt * 3 nt-pairs = 1296 waves -> 162 blocks.
// ---------------------------------------------------------------------------
__global__ __launch_bounds__(256) void qk_wmma(const float* __restrict__ q,
                                               const float* __restrict__ kc,
                                               float* __restrict__ S) {
  const int wave = threadIdx.x >> 5;
  const int lane = threadIdx.x & 31;
  const int tile = blockIdx.x * 8 + wave;   // 0..1295
  const int bp = tile / 18;                 // 0..71
  const int mn = tile - bp * 18;
  const int mt = mn / 3;
  const int ntp = mn - mt * 3;
  const int b  = bp / 9;
  const int p  = bp - b * 9;
  const int ph = p / 3;
  const int pw = p - ph * 3;
  const int hl  = lane & 15;
  const int kh2 = (lane >> 4) << 1;

  const int c1  = (mt << 4) + hl;
  const int c2a = (ntp << 5) + hl;
  const float* qb  = q  + ((size_t)b * cC + c1)  * cHW + pw;
  const float* kba = kc + ((size_t)b * cC + c2a) * cHW + pw;
  const float* kbb = kba + (size_t)16 * cHW;

  v8f acc0 = {}, acc1 = {};
  for (int li = 0; li < 32; ++li) {
    const int base = (3 * li + ph) * cW + 3 * kh2;
#pragma unroll 4
    for (int lj0 = 0; lj0 < 32; lj0 += 4) {
      const int o = base + 3 * lj0;         // slots at o and o+3
      v2f a, b0, b1;
      a[0]  = qb[o];   a[1]  = qb[o + 3];
      b0[0] = kba[o];  b0[1] = kba[o + 3];
      b1[0] = kbb[o];  b1[1] = kbb[o + 3];
      acc0 = wmma_f32(a, b0, acc0);
      acc1 = wmma_f32(a, b1, acc1);
    }
  }

  // Scatter into softmax-row layout S[b, c2, c1*9 + p].
#pragma unroll
  for (int rr = 0; rr < 8; ++rr) {
    const int mm = (mt << 4) + rr + ((lane >> 4) << 3);
    const int nn0 = (ntp << 5) + hl;
    S[((size_t)b * cC + nn0) * cM + (size_t)mm * 9 + p] = acc0[rr];
    S[((size_t)b * cC + nn0 + 16) * cM + (size_t)mm * 9 + p] = acc1[rr];
  }
}

// ---------------------------------------------------------------------------
// Row softmax over 864 entries (scaled by 1/sqrt(864)). 768 rows.
// ---------------------------------------------------------------------------
__global__ __launch_bounds__(256) void softmax_rows(float* __restrict__ S) {
  __shared__ float red[256];
  float* p = S + (size_t)blockIdx.x * cM;
  const int t = threadIdx.x;

  float mx = -INFINITY;
  for (int i = t; i < cM; i += 256) mx = fmaxf(mx, p[i] * cSCALE);
  red[t] = mx;
  __syncthreads();
  for (int s = 128; s > 0; s >>= 1) {
    if (t < s) red[t] = fmaxf(red[t], red[t + s]);
    __syncthreads();
  }
  mx = red[0];
  __syncthreads();

  float sum = 0.0f;
  for (int i = t; i < cM; i += 256) {
    const float e = __expf(p[i] * cSCALE - mx);
    p[i] = e;
    sum += e;
  }
  red[t] = sum;
  __syncthreads();
  for (int s = 128; s > 0; s >>= 1) {
    if (t < s) red[t] += red[t + s];
    __syncthreads();
  }
  const float inv = 1.0f / red[0];
  for (int i = t; i < cM; i += 256) p[i] *= inv;
}

// ---------------------------------------------------------------------------
// BatchNorm2d train-mode stats (biased var over B,H,W) per channel.
// ---------------------------------------------------------------------------
__global__ __launch_bounds__(256) void bn_stats(const float* __restrict__ x,
                                                float* __restrict__ mu,
                                                float* __restrict__ rs) {
  __shared__ float s1[256], s2[256];
  const int c = blockIdx.x;
  const int t = threadIdx.x;
  float a = 0.0f, q = 0.0f;
  for (int i = t; i < cNPIX; i += 256) {
    const int b = i / cHW;
    const int s = i - b * cHW;
    const float v = x[((size_t)b * cC + c) * cHW + s];
    a += v;
    q += v * v;
  }
  s1[t] = a;
  s2[t] = q;
  __syncthreads();
  for (int s = 128; s > 0; s >>= 1) {
    if (t < s) { s1[t] += s1[t + s]; s2[t] += s2[t + s]; }
    __syncthreads();
  }
  if (t == 0) {
    const float m   = s1[0] * (1.0f / cNPIX);
    const float var = s2[0] * (1.0f / cNPIX) - m * m;
    mu[c] = m;
    rs[c] = rsqrtf(var + 1e-5f);
  }
}

// ---------------------------------------------------------------------------
// xcat = concat(x1, x2) along channels -> written to second output slot.
// ---------------------------------------------------------------------------
__global__ __launch_bounds__(256) void concat2(const float* __restrict__ x1,
                                               const float* __restrict__ x2,
                                               float* __restrict__ xcat) {
  const size_t idx = (size_t)blockIdx.x * 256 + threadIdx.x;
  const size_t hw = idx % cHW;
  const size_t bc = idx / cHW;
  const int ch = (int)(bc % (2 * cC));
  const int b  = (int)(bc / (2 * cC));
  float v;
  if (ch < cC) v = x1[((size_t)b * cC + ch) * cHW + hw];
  else         v = x2[((size_t)b * cC + (ch - cC)) * cHW + hw];
  xcat[idx] = v;
}

// ---------------------------------------------------------------------------
// out1 = xcat + concat( bn(y1)*bn(y2), bn(y3) )
// st = [mu1|rs1|mu2|rs2|mu3|rs3], 96 floats each.
// ---------------------------------------------------------------------------
__global__ __launch_bounds__(256) void fuse_out(
    const float* __restrict__ xcat, const float* __restrict__ y1,
    const float* __restrict__ y2, const float* __restrict__ y3,
    const float* __restrict__ st, float* __restrict__ out1) {
  const size_t idx = (size_t)blockIdx.x * 256 + threadIdx.x;
  const size_t hw = idx % cHW;
  const size_t bc = idx / cHW;
  const int ch = (int)(bc % (2 * cC));
  const int b  = (int)(bc / (2 * cC));
  float yv;
  if (ch < cC) {
    const size_t o = ((size_t)b * cC + ch) * cHW + hw;
    const float g1 = (y1[o] - st[ch])       * st[96 + ch];
    const float g2 = (y2[o] - st[192 + ch]) * st[288 + ch];
    yv = g1 * g2;
  } else {
    const int c2 = ch - cC;
    const size_t o = ((size_t)b * cC + c2) * cHW + hw;
    yv = (y3[o] - st[384 + c2]) * st[480 + c2];
  }
  out1[idx] = xcat[idx] + yv;
}

// ---------------------------------------------------------------------------
extern "C" void kernel_launch(void* const* d_in, const int* in_sizes, int n_in,
                              void* d_out, int out_size, void* d_ws,
                              size_t ws_size, hipStream_t stream) {
  (void)in_sizes; (void)n_in; (void)out_size; (void)ws_size;
  const float* x1  = (const float*)d_in[0];
  const float* x2  = (const float*)d_in[1];
  const float* w1  = (const float*)d_in[2];
  const float* w2  = (const float*)d_in[3];
  const float* wa1 = (const float*)d_in[4];
  const float* wa2 = (const float*)d_in[5];
  const float* wa3 = (const float*)d_in[6];

  float* out = (float*)d_out;
  const size_t HALF = (size_t)cB * 2 * cC * cHW;  // 14,155,776
  float* xcat = out + HALF;                        // second tuple element = x

  const size_t BIG = (size_t)cB * cC * cHW;        // 7,077,888 floats
  float* W0 = (float*)d_ws;                        // q -> v
  float* W1 = W0 + BIG;                            // k -> y1
  float* W2 = W1 + BIG;                            // y2
  float* W3 = W2 + BIG;                            // y3
  float* S  = W3 + BIG;                            // attn rows [8*96, 864]
  float* WT = S + (size_t)cB * cC * cM;            // transposed weights
  float* ZB = WT + (size_t)cB * cC * cM;           // zero page
  float* st = ZB + cZB;                            // 6*96 stats

  const int convBlocks = (6 * (cNPIX / 32)) / 8;   // 1728
  const int qkBlocks   = (72 * 18) / 8;            // 162
  const int elemBlocks = (int)(HALF / 256);        // 55296
  const int zbBlocks   = (cZB + 255) / 256;
  const int ro2cBlocks = (cC * 2 * cC * 9 + 255) / 256;      // 96*1728
  const int ro1cBlocks = (cC * cC * 9 + 255) / 256;          // 96*864
  const int rodBlocks  = (cB * cC * cC * 9 + 255) / 256;     // 8*96*864

  fill_zero<<<zbBlocks, 256, 0, stream>>>(ZB, cZB);
  // x = concat(x1,x2): conv input and second output.
  concat2<<<elemBlocks, 256, 0, stream>>>(x1, x2, xcat);

  // Attention branch: q, k -> QK GEMM -> softmax -> v -> PV as dynamic conv.
  reorder_w<<<ro2cBlocks, 256, 0, stream>>>(wa1, WT, 2 * cC, 1);
  conv3x3_wmma<<<convBlocks, 256, 0, stream>>>(xcat, WT, W0, ZB, 2 * cC, 0); // q
  reorder_w<<<ro2cBlocks, 256, 0, stream>>>(wa2, WT, 2 * cC, 1);
  conv3x3_wmma<<<convBlocks, 256, 0, stream>>>(xcat, WT, W1, ZB, 2 * cC, 0); // k
  qk_wmma<<<qkBlocks, 256, 0, stream>>>(W0, W1, S);
  softmax_rows<<<cB * cC, 256, 0, stream>>>(S);
  reorder_w<<<ro2cBlocks, 256, 0, stream>>>(wa3, WT, 2 * cC, 1);
  conv3x3_wmma<<<convBlocks, 256, 0, stream>>>(xcat, WT, W0, ZB, 2 * cC, 0); // v
  // Gating branch convs (reuse freed buffers).
  reorder_w<<<ro1cBlocks, 256, 0, stream>>>(w1, WT, cC, 1);
  conv3x3_wmma<<<convBlocks, 256, 0, stream>>>(x1, WT, W1, ZB, cC, 0);       // y1
  reorder_w<<<ro1cBlocks, 256, 0, stream>>>(w2, WT, cC, 1);
  conv3x3_wmma<<<convBlocks, 256, 0, stream>>>(x2, WT, W2, ZB, cC, 0);       // y2
  // PV: y3[b] = conv3x3(v[b], A[b]) with per-batch softmaxed weights.
  reorder_w<<<rodBlocks, 256, 0, stream>>>(S, WT, cC, cB);
  conv3x3_wmma<<<convBlocks, 256, 0, stream>>>(W0, WT, W3, ZB, cC, cC * cM); // y3

  // BatchNorm stats.
  bn_stats<<<cC, 256, 0, stream>>>(W1, st + 0,   st + 96);
  bn_stats<<<cC, 256, 0, stream>>>(W2, st + 192, st + 288);
  bn_stats<<<cC, 256, 0, stream>>>(W3, st + 384, st + 480);

  // Final fusion: out1 = x + concat(bn(y1)*bn(y2), bn(y3)).
  fuse_out<<<elemBlocks, 256, 0, stream>>>(xcat, W1, W2, W3, st, out);
}